// GIN_39187281609360
// MI455X (gfx1250) — compile-verified
//
#include <hip/hip_runtime.h>

// ---------------------------------------------------------------------------
// GIN forward for MI455X (gfx1250).
//  - All MLP GEMMs on the CDNA5 matrix pipe via V_WMMA_F32_16X16X4_F32
//    (exact fp32; workload is memory/atomic bound so low precision buys zero).
//  - Node MLP: 128-thread blocks (4 wave32), 64 nodes/block = 4 row-tiles.
//    Per layer each wave loads its 16-column B slice ONCE into registers
//    (fully unrolled, clause-batched) and sweeps 4 row-tiles -> 4x weight
//    traffic amortization and long WMMA chains.
//  - Edge scatter / pooling: float4 gathers + global_atomic_add_f32 (L2).
// ---------------------------------------------------------------------------

typedef __attribute__((ext_vector_type(2))) float v2f;
typedef __attribute__((ext_vector_type(8))) float v8f;

#define TILE_STR 68  // LDS row stride (floats): 68 % 64 == 4 -> conflict-free

__device__ __forceinline__ v8f wmma_f32_step(v2f a, v2f b, v8f c) {
  // 8 args: (neg_a, A, neg_b, B, c_mod, C, reuse_a, reuse_b)
  return __builtin_amdgcn_wmma_f32_16x16x4_f32(false, a, false, b, (short)0, c,
                                               false, false);
}

// B 4x16 f32 fragment layout (ISA 7.12.2): col = lane&15, half = lane>>4;
// vgpr0 = row (k0+2h), vgpr1 = row (k0+2h+1). Load all K/4 fragments of the
// wave's 16-column W slice into registers (fully unrolled -> load clause).
template <int K, int NW>
__device__ __forceinline__ void load_bfrags(const float* __restrict__ W,
                                            int col, v2f* bf) {
  const int half = (threadIdx.x & 31) >> 4;
#pragma unroll
  for (int k = 0; k < K / 4; ++k) {
    bf[k].x = W[(4 * k + 2 * half) * NW + col];
    bf[k].y = W[(4 * k + 2 * half + 1) * NW + col];
  }
}

// A 16x4 f32 fragment from LDS row-tile; chained WMMA over K.
template <int K>
__device__ __forceinline__ v8f gemm_rowtile(const float* Xs_rt, const v2f* bf) {
  const int lane = threadIdx.x & 31;
  const int half = lane >> 4;
  const int m = lane & 15;
  v8f acc = {0.f, 0.f, 0.f, 0.f, 0.f, 0.f, 0.f, 0.f};
#pragma unroll
  for (int k = 0; k < K / 4; ++k) {
    v2f a;
    a.x = Xs_rt[m * TILE_STR + 4 * k + 2 * half];
    a.y = Xs_rt[m * TILE_STR + 4 * k + 2 * half + 1];
    acc = wmma_f32_step(a, bf[k], acc);
  }
  return acc;
}

// D layout (f32 16x16): vgpr r, lanes 0-15 -> row r, lanes 16-31 -> row r+8.
__device__ __forceinline__ void store_tile_lds(float* Ys_rt, v8f acc, int col,
                                               float s, float t) {
  const int half = (threadIdx.x & 31) >> 4;
#pragma unroll
  for (int r = 0; r < 8; ++r) {
    float v = acc[r] * s + t;
    Ys_rt[(r + 8 * half) * TILE_STR + col] = fmaxf(v, 0.f);
  }
}

// --------------------------------- kernels ---------------------------------

__global__ void gin_zero(float* __restrict__ p, long long n) {
  long long i = (long long)blockIdx.x * blockDim.x + threadIdx.x;
  const long long stride = (long long)gridDim.x * blockDim.x;
  for (; i < n; i += stride) p[i] = 0.f;
}

// Per (edge, 4-channel group): e = edge_attr @ We + be ; m = relu(h[src]+e) ;
// agg[dst] += m  (f32 atomics into L2-resident agg).
template <int CIN>
__global__ void gin_edge(const float* __restrict__ h,
                         const int* __restrict__ ei,   // [2, E]
                         const float* __restrict__ ea, // [E, 2]
                         const float* __restrict__ We, // [2, CIN]
                         const float* __restrict__ be, // [CIN]
                         float* __restrict__ agg,      // [N, CIN]
                         int n_edges) {
  constexpr int G = CIN / 4;
  const long long total = (long long)n_edges * G;
  long long i = (long long)blockIdx.x * blockDim.x + threadIdx.x;
  const long long stride = (long long)gridDim.x * blockDim.x;
  for (; i < total; i += stride) {
    const int e = (int)(i / G);
    const int c = (int)(i - (long long)e * G) * 4;
    const int s = ei[e];
    const int d = ei[n_edges + e];
    const float ea0 = ea[2 * e];
    const float ea1 = ea[2 * e + 1];
    const float4 hv = *(const float4*)(h + (long long)s * CIN + c);
    float* ap = agg + (long long)d * CIN + c;
    float m0 = fmaxf(hv.x + ea0 * We[c + 0] + ea1 * We[CIN + c + 0] + be[c + 0], 0.f);
    float m1 = fmaxf(hv.y + ea0 * We[c + 1] + ea1 * We[CIN + c + 1] + be[c + 1], 0.f);
    float m2 = fmaxf(hv.z + ea0 * We[c + 2] + ea1 * We[CIN + c + 2] + be[c + 2], 0.f);
    float m3 = fmaxf(hv.w + ea0 * We[c + 3] + ea1 * We[CIN + c + 3] + be[c + 3], 0.f);
    atomicAdd(ap + 0, m0);
    atomicAdd(ap + 1, m1);
    atomicAdd(ap + 2, m2);
    atomicAdd(ap + 3, m3);
  }
}

// global_add_pool: pooled[batch[n]][c..c+3] += r[n][c..c+3]
template <int C>
__global__ void gin_pool(const float* __restrict__ r,
                         const int* __restrict__ batch,
                         float* __restrict__ pooled, int n_nodes) {
  constexpr int G = C / 4;
  const long long total = (long long)n_nodes * G;
  long long i = (long long)blockIdx.x * blockDim.x + threadIdx.x;
  const long long stride = (long long)gridDim.x * blockDim.x;
  for (; i < total; i += stride) {
    const int n = (int)(i / G);
    const int c = (int)(i - (long long)n * G) * 4;
    const float4 v = *(const float4*)(r + (long long)n * C + c);
    float* pp = pooled + (long long)batch[n] * C + c;
    atomicAdd(pp + 0, v.x);
    atomicAdd(pp + 1, v.y);
    atomicAdd(pp + 2, v.z);
    atomicAdd(pp + 3, v.w);
  }
}

// h_new = relu( L3( relu(BN2(L2( relu(BN1(L1( (1+eps)*h + agg )))))) ) )
// Block: 128 threads = 4 wave32; 64 nodes/block (4 row-tiles of 16).
template <int CIN>
__global__ void gin_node_mlp(const float* __restrict__ hin,  // [N, CIN]
                             const float* __restrict__ agg,  // [N, CIN]
                             const float* __restrict__ epsp, // scalar
                             const float* __restrict__ W1, const float* __restrict__ b1,
                             const float* __restrict__ beta1, const float* __restrict__ g1,
                             const float* __restrict__ W2, const float* __restrict__ b2,
                             const float* __restrict__ beta2, const float* __restrict__ g2,
                             const float* __restrict__ W3, const float* __restrict__ b3,
                             float* __restrict__ hout, // [N, 64]
                             int n_nodes) {
  __shared__ float Xs[64 * TILE_STR];
  __shared__ float Ys[64 * TILE_STR];
  const int tid = threadIdx.x;
  const int lane = tid & 31;
  const int wave = tid >> 5;
  const long long base = (long long)blockIdx.x * 64;
  const float opE = 1.0f + epsp[0];

  // Stage fused input tile: x0 = (1+eps)*h + agg
#pragma unroll
  for (int i = tid; i < 64 * CIN; i += 128) {
    const int row = i / CIN;
    const int c = i % CIN;
    const long long node = base + row;
    float v = 0.f;
    if (node < n_nodes) {
      const long long o = node * CIN + c;
      v = opE * hin[o] + agg[o];
    }
    Xs[row * TILE_STR + c] = v;
  }
  __syncthreads();

  const float bns = rsqrtf(1.0f + 1e-5f);  // eval-mode BN (running var = 1)
  const int col = wave * 16 + (lane & 15);
  v2f bf[16];

  // Layer 1: CIN -> 64, BN-fold + ReLU
  load_bfrags<CIN, 64>(W1, col, bf);
  {
    const float s = g1[col] * bns;
    const float t = b1[col] * s + beta1[col];
#pragma unroll
    for (int rt = 0; rt < 4; ++rt) {
      v8f acc = gemm_rowtile<CIN>(Xs + rt * 16 * TILE_STR, bf);
      store_tile_lds(Ys + rt * 16 * TILE_STR, acc, col, s, t);
    }
  }
  __syncthreads();

  // Layer 2: 64 -> 64, BN-fold + ReLU
  load_bfrags<64, 64>(W2, col, bf);
  {
    const float s = g2[col] * bns;
    const float t = b2[col] * s + beta2[col];
#pragma unroll
    for (int rt = 0; rt < 4; ++rt) {
      v8f acc = gemm_rowtile<64>(Ys + rt * 16 * TILE_STR, bf);
      store_tile_lds(Xs + rt * 16 * TILE_STR, acc, col, s, t);
    }
  }
  __syncthreads();

  // Layer 3: 64 -> 64, + bias, outer ReLU, store h_new
  load_bfrags<64, 64>(W3, col, bf);
  {
    const float t = b3[col];
    const int half = lane >> 4;
#pragma unroll
    for (int rt = 0; rt < 4; ++rt) {
      v8f acc = gemm_rowtile<64>(Xs + rt * 16 * TILE_STR, bf);
#pragma unroll
      for (int r = 0; r < 8; ++r) {
        const long long node = base + rt * 16 + r + 8 * half;
        if (node < n_nodes)
          hout[node * 64 + col] = fmaxf(acc[r] + t, 0.f);
      }
    }
  }
}

// Readout MLP on pooled [64, CIN] -> accumulate into out [64, 16].
// Single block, 128 threads (4 waves), 4 row-tiles.
template <int CIN>
__global__ void gin_readout(const float* __restrict__ pooled,
                            const float* __restrict__ W1, const float* __restrict__ b1,
                            const float* __restrict__ beta1, const float* __restrict__ g1,
                            const float* __restrict__ W2, const float* __restrict__ b2,
                            const float* __restrict__ beta2, const float* __restrict__ g2,
                            const float* __restrict__ W3, const float* __restrict__ b3,
                            float* __restrict__ out) {
  __shared__ float Xs[64 * TILE_STR];
  __shared__ float Ys[64 * TILE_STR];
  const int tid = threadIdx.x;
  const int lane = tid & 31;
  const int wave = tid >> 5;

#pragma unroll
  for (int i = tid; i < 64 * CIN; i += 128) {
    const int row = i / CIN;
    const int c = i % CIN;
    Xs[row * TILE_STR + c] = pooled[row * CIN + c];
  }
  __syncthreads();

  const float bns = rsqrtf(1.0f + 1e-5f);
  const int col = wave * 16 + (lane & 15);
  v2f bf[16];

  load_bfrags<CIN, 64>(W1, col, bf);
  {
    const float s = g1[col] * bns;
    const float t = b1[col] * s + beta1[col];
#pragma unroll
    for (int rt = 0; rt < 4; ++rt) {
      v8f acc = gemm_rowtile<CIN>(Xs + rt * 16 * TILE_STR, bf);
      store_tile_lds(Ys + rt * 16 * TILE_STR, acc, col, s, t);
    }
  }
  __syncthreads();

  load_bfrags<64, 64>(W2, col, bf);
  {
    const float s = g2[col] * bns;
    const float t = b2[col] * s + beta2[col];
#pragma unroll
    for (int rt = 0; rt < 4; ++rt) {
      v8f acc = gemm_rowtile<64>(Ys + rt * 16 * TILE_STR, bf);
      store_tile_lds(Xs + rt * 16 * TILE_STR, acc, col, s, t);
    }
  }
  __syncthreads();

  // Final 64 -> 16: wave 0 only (uniform per-wave branch; EXEC all-ones inside).
  if (wave == 0) {
    const int oc = lane & 15;
    const int half = lane >> 4;
    v2f bf3[16];
    load_bfrags<64, 16>(W3, oc, bf3);
    const float t = b3[oc];
#pragma unroll
    for (int rt = 0; rt < 4; ++rt) {
      v8f acc = gemm_rowtile<64>(Xs + rt * 16 * TILE_STR, bf3);
#pragma unroll
      for (int r = 0; r < 8; ++r) {
        const int row = rt * 16 + r + 8 * half;
        out[row * 16 + oc] += acc[r] + t;  // sequential readout launches: safe RMW
      }
    }
  }
}

// -------------------------------- launcher ---------------------------------

extern "C" void kernel_launch(void* const* d_in, const int* in_sizes, int n_in,
                              void* d_out, int out_size, void* d_ws,
                              size_t ws_size, hipStream_t stream) {
  const float* x     = (const float*)d_in[0];
  const int*   ei    = (const int*)d_in[1];
  const float* ea    = (const float*)d_in[2];
  const int*   batch = (const int*)d_in[3];
  // d_in[4] = batch_size (64)
  const int n_nodes = in_sizes[0] / 32;  // IN_C = 32
  const int n_edges = in_sizes[1] / 2;

  // params leaves in pytree order (sorted dict keys):
  //   convs[i]: We, be, eps, mlp{ W,b,beta,g | W,b,beta,g | W,b }   -> 13 leaves
  //   readouts[r]: { W,b,beta,g | W,b,beta,g | W,b }                -> 10 leaves
  int p = 5;
  const float *cWe[4], *cbe[4], *ceps[4], *cm[4][10];
  for (int i = 0; i < 4; ++i) {
    cWe[i]  = (const float*)d_in[p++];
    cbe[i]  = (const float*)d_in[p++];
    ceps[i] = (const float*)d_in[p++];
    for (int j = 0; j < 10; ++j) cm[i][j] = (const float*)d_in[p++];
  }
  const float* ro[5][10];
  for (int r = 0; r < 5; ++r)
    for (int j = 0; j < 10; ++j) ro[r][j] = (const float*)d_in[p++];

  // workspace layout (floats)
  float* ws = (float*)d_ws;
  const long long NH = (long long)n_nodes * 64;
  float* hA     = ws;
  float* hB     = ws + NH;
  float* agg    = ws + 2 * NH;
  float* pooled = ws + 3 * NH;  // 5 reps x 4096 floats

  gin_zero<<<80, 256, 0, stream>>>(pooled, 5LL * 4096);
  gin_pool<32><<<1024, 256, 0, stream>>>(x, batch, pooled, n_nodes);

  const float* hcur = x;
  float* bufs[2] = {hA, hB};
  const int nblk = (n_nodes + 63) / 64;
  for (int l = 0; l < 4; ++l) {
    float* hnext = bufs[l & 1];
    if (l == 0) {
      gin_zero<<<1024, 256, 0, stream>>>(agg, (long long)n_nodes * 32);
      gin_edge<32><<<2048, 256, 0, stream>>>(hcur, ei, ea, cWe[l], cbe[l], agg,
                                             n_edges);
      gin_node_mlp<32><<<nblk, 128, 0, stream>>>(
          hcur, agg, ceps[l],
          cm[l][0], cm[l][1], cm[l][2], cm[l][3],
          cm[l][4], cm[l][5], cm[l][6], cm[l][7],
          cm[l][8], cm[l][9], hnext, n_nodes);
    } else {
      gin_zero<<<1024, 256, 0, stream>>>(agg, (long long)n_nodes * 64);
      gin_edge<64><<<2048, 256, 0, stream>>>(hcur, ei, ea, cWe[l], cbe[l], agg,
                                             n_edges);
      gin_node_mlp<64><<<nblk, 128, 0, stream>>>(
          hcur, agg, ceps[l],
          cm[l][0], cm[l][1], cm[l][2], cm[l][3],
          cm[l][4], cm[l][5], cm[l][6], cm[l][7],
          cm[l][8], cm[l][9], hnext, n_nodes);
    }
    gin_pool<64><<<1024, 256, 0, stream>>>(hnext, batch, pooled + (l + 1) * 4096,
                                           n_nodes);
    hcur = hnext;
  }

  float* out = (float*)d_out;
  gin_zero<<<4, 256, 0, stream>>>(out, 64LL * 16);
  for (int r = 0; r < 5; ++r) {
    if (r == 0) {
      gin_readout<32><<<1, 128, 0, stream>>>(
          pooled + r * 4096,
          ro[r][0], ro[r][1], ro[r][2], ro[r][3],
          ro[r][4], ro[r][5], ro[r][6], ro[r][7],
          ro[r][8], ro[r][9], out);
    } else {
      gin_readout<64><<<1, 128, 0, stream>>>(
          pooled + r * 4096,
          ro[r][0], ro[r][1], ro[r][2], ro[r][3],
          ro[r][4], ro[r][5], ro[r][6], ro[r][7],
          ro[r][8], ro[r][9], out);
    }
  }
}